// Overall_model_stepgame_47931835023577
// MI455X (gfx1250) — compile-verified
//
#include <hip/hip_runtime.h>

// Problem constants (from reference): B=64, N=64, E=128, D=128, PF=128
#define BB 64
#define NN 64
#define EE 128
#define DD 128
#define PFF 128

typedef float v8f __attribute__((ext_vector_type(8)));
typedef float v2f __attribute__((ext_vector_type(2)));

// ---- f32 WMMA 16x16x4:  D(16x16 f32, v8f) = A(16x4 f32, v2f) * B(4x16 f32, v2f) + C
static __device__ __forceinline__ v8f wmma4(v2f a, v2f b, v8f c) {
  return __builtin_amdgcn_wmma_f32_16x16x4_f32(
      /*neg_a=*/false, a, /*neg_b=*/false, b,
      /*c_mod=*/(short)0, c, /*reuse_a=*/false, /*reuse_b=*/false);
}

static __device__ __forceinline__ v8f splat8(float x) {
  v8f r;
#pragma unroll
  for (int i = 0; i < 8; ++i) r[i] = x;
  return r;
}

// For one K-pair (k,k+1): broadcast B fragments for all 8 N-tiles of a 128-wide
// output and accumulate 8 WMMAs.  pw points at a (K x 128) row-major weight.
static __device__ __forceinline__ void wmma_row(const float* __restrict__ pw,
                                                int k, int col, v2f a, v8f acc[8]) {
#pragma unroll
  for (int nt = 0; nt < 8; ++nt) {
    v2f bb;
    bb.x = pw[k * DD + nt * 16 + col];
    bb.y = pw[(k + 1) * DD + nt * 16 + col];
    acc[nt] = wmma4(a, bb, acc[nt]);
  }
}

// ----------------------------------------------------------------------------
__global__ void zero_kernel(float* __restrict__ p, int n) {
  int i = blockIdx.x * blockDim.x + threadIdx.x;
  if (i < n) p[i] = 0.0f;
}

// ----------------------------------------------------------------------------
// Edge pipeline: per wave = one 16-edge tile of one batch.
//   seq = [h_src | edge | h_dst]  (16 x 384)
//   x1 = seq@W0+b0 ; x2 = x1@W1+b1 ; fill = x2@W2+b2
//   s_e = dot(h_dst_e, tgt_b)
//   R[b, src_e, :] += fill_e * s_e        (atomic f32 scatter-add)
__global__ __launch_bounds__(128) void edge_kernel(
    const float* __restrict__ nf, const float* __restrict__ ef,
    const float* __restrict__ tgt, const int* __restrict__ ei,
    const float* __restrict__ W0, const float* __restrict__ b0,
    const float* __restrict__ W1, const float* __restrict__ b1,
    const float* __restrict__ W2, const float* __restrict__ b2,
    float* __restrict__ R) {
  __shared__ __attribute__((aligned(16))) float xbuf[4][16 * DD];
  __shared__ float sbuf[4][16];
  __shared__ int srcbuf[4][16];

  const int lane = threadIdx.x & 31;
  const int wv = threadIdx.x >> 5;
  const int hi = lane >> 4;   // 0/1 half of the wave
  const int col = lane & 15;  // fragment row (A) / col (B,C)

  const int tile = blockIdx.x * 4 + wv;  // 512 tiles total
  const int b = tile >> 3;               // batch
  const int et = tile & 7;               // edge tile in batch
  const int e = et * 16 + col;           // this lane's edge row

  const int srcm = ei[(b * 2 + 0) * EE + e];
  const int dstm = ei[(b * 2 + 1) * EE + e];

  const float* aseg0 = nf + (size_t)(b * NN + srcm) * DD;  // h_src row
  const float* aseg1 = ef + (size_t)(b * EE + e) * DD;     // edge row
  const float* aseg2 = nf + (size_t)(b * NN + dstm) * DD;  // h_dst row

  // s = dot(h_dst, target): each half-wave sums 64 elements, combine via shfl.
  {
    const float* tg = tgt + b * DD;
    float p = 0.0f;
#pragma unroll 8
    for (int j = 0; j < 64; ++j) p += aseg2[hi * 64 + j] * tg[hi * 64 + j];
    p += __shfl_xor(p, 16);
    if (hi == 0) {
      sbuf[wv][col] = p;
      srcbuf[wv][col] = srcm;
    }
  }

  float* xb = xbuf[wv];
  v8f acc[8];

  // ---- GEMM1: x1 = seq @ W0 + b0   (K = 384 over 3 segments) ----
#pragma unroll
  for (int nt = 0; nt < 8; ++nt) acc[nt] = splat8(b0[nt * 16 + col]);
  const float* asegs[3] = {aseg0, aseg1, aseg2};
#pragma unroll 1
  for (int seg = 0; seg < 3; ++seg) {
    const float* pa = asegs[seg];
    const float* pw = W0 + seg * DD * DD;
#pragma unroll 1
    for (int t = 0; t < 32; ++t) {
      const int k = t * 4 + hi * 2;
      v2f a = *(const v2f*)(pa + k);
      wmma_row(pw, k, col, a, acc);
    }
  }
  // C-layout -> row-major LDS tile
#pragma unroll
  for (int nt = 0; nt < 8; ++nt)
#pragma unroll
    for (int v = 0; v < 8; ++v)
      xb[(v + 8 * hi) * DD + nt * 16 + col] = acc[nt][v];
  __syncthreads();

  // ---- GEMM2: x2 = x1 @ W1 + b1  (K = 128) ----
#pragma unroll
  for (int nt = 0; nt < 8; ++nt) acc[nt] = splat8(b1[nt * 16 + col]);
#pragma unroll 1
  for (int t = 0; t < 32; ++t) {
    const int k = t * 4 + hi * 2;
    v2f a = *(const v2f*)(xb + col * DD + k);
    wmma_row(W1, k, col, a, acc);
  }
  __syncthreads();
#pragma unroll
  for (int nt = 0; nt < 8; ++nt)
#pragma unroll
    for (int v = 0; v < 8; ++v)
      xb[(v + 8 * hi) * DD + nt * 16 + col] = acc[nt][v];
  __syncthreads();

  // ---- GEMM3: fill = x2 @ W2 + b2  (K = 128) ----
#pragma unroll
  for (int nt = 0; nt < 8; ++nt) acc[nt] = splat8(b2[nt * 16 + col]);
#pragma unroll 1
  for (int t = 0; t < 32; ++t) {
    const int k = t * 4 + hi * 2;
    v2f a = *(const v2f*)(xb + col * DD + k);
    wmma_row(W2, k, col, a, acc);
  }

  // scale by s(row) and scatter-add into R[b, src(row), p]
#pragma unroll
  for (int nt = 0; nt < 8; ++nt) {
    const int p = nt * 16 + col;
#pragma unroll
    for (int v = 0; v < 8; ++v) {
      const int m2 = v + 8 * hi;  // edge row within tile
      const float val = acc[nt][v] * sbuf[wv][m2];
      atomicAdd(&R[(size_t)(b * NN + srcbuf[wv][m2]) * PFF + p], val);
    }
  }
}

// ----------------------------------------------------------------------------
// Node pipeline: per wave = one 16-node tile of one batch.
//   r = LayerNorm(R[b,n,:])*gamma+beta
//   inp = [nf | r | tgt]  (16 x 384)
//   y = ((inp@V0+c0)@V1+c1)@V2+c2  -> out[b,n,:]
__global__ __launch_bounds__(128) void node_kernel(
    const float* __restrict__ nf, const float* __restrict__ tgt,
    const float* __restrict__ gamma, const float* __restrict__ beta,
    const float* __restrict__ V0, const float* __restrict__ c0,
    const float* __restrict__ V1, const float* __restrict__ c1,
    const float* __restrict__ V2, const float* __restrict__ c2,
    const float* __restrict__ R, float* __restrict__ out) {
  __shared__ __attribute__((aligned(16))) float xbuf[4][16 * DD];

  const int lane = threadIdx.x & 31;
  const int wv = threadIdx.x >> 5;
  const int hi = lane >> 4;
  const int col = lane & 15;

  const int tile = blockIdx.x * 4 + wv;  // 256 tiles total
  const int b = tile >> 2;
  const int ntb = (tile & 3) * 16;
  const int n = ntb + col;  // this lane's node row

  const float* rrow = R + (size_t)(b * NN + n) * PFF;

  // LayerNorm stats over PF=128 (biased variance, matching jnp.var)
  float s = 0.0f, ss = 0.0f;
#pragma unroll 8
  for (int j = 0; j < 64; ++j) {
    float x = rrow[hi * 64 + j];
    s += x;
    ss += x * x;
  }
  s += __shfl_xor(s, 16);
  ss += __shfl_xor(ss, 16);
  const float mean = s * (1.0f / 128.0f);
  const float var = ss * (1.0f / 128.0f) - mean * mean;
  const float rstd = rsqrtf(var + 1e-5f);

  float* xb = xbuf[wv];
  v8f acc[8];

  // ---- GEMM V0: y1 = inp @ V0 + c0  (K = 384, 3 segments, A built on the fly)
#pragma unroll
  for (int nt = 0; nt < 8; ++nt) acc[nt] = splat8(c0[nt * 16 + col]);
  {
    const float* pa = nf + (size_t)(b * NN + n) * DD;
#pragma unroll 1
    for (int t = 0; t < 32; ++t) {
      const int k = t * 4 + hi * 2;
      v2f a = *(const v2f*)(pa + k);
      wmma_row(V0, k, col, a, acc);
    }
#pragma unroll 1
    for (int t = 0; t < 32; ++t) {
      const int k = t * 4 + hi * 2;
      v2f a;
      a.x = (rrow[k] - mean) * rstd * gamma[k] + beta[k];
      a.y = (rrow[k + 1] - mean) * rstd * gamma[k + 1] + beta[k + 1];
      wmma_row(V0 + DD * DD, k, col, a, acc);
    }
    const float* tg = tgt + b * DD;
#pragma unroll 1
    for (int t = 0; t < 32; ++t) {
      const int k = t * 4 + hi * 2;
      v2f a = *(const v2f*)(tg + k);
      wmma_row(V0 + 2 * DD * DD, k, col, a, acc);
    }
  }
#pragma unroll
  for (int nt = 0; nt < 8; ++nt)
#pragma unroll
    for (int v = 0; v < 8; ++v)
      xb[(v + 8 * hi) * DD + nt * 16 + col] = acc[nt][v];
  __syncthreads();

  // ---- GEMM V1 ----
#pragma unroll
  for (int nt = 0; nt < 8; ++nt) acc[nt] = splat8(c1[nt * 16 + col]);
#pragma unroll 1
  for (int t = 0; t < 32; ++t) {
    const int k = t * 4 + hi * 2;
    v2f a = *(const v2f*)(xb + col * DD + k);
    wmma_row(V1, k, col, a, acc);
  }
  __syncthreads();
#pragma unroll
  for (int nt = 0; nt < 8; ++nt)
#pragma unroll
    for (int v = 0; v < 8; ++v)
      xb[(v + 8 * hi) * DD + nt * 16 + col] = acc[nt][v];
  __syncthreads();

  // ---- GEMM V2 -> out ----
#pragma unroll
  for (int nt = 0; nt < 8; ++nt) acc[nt] = splat8(c2[nt * 16 + col]);
#pragma unroll 1
  for (int t = 0; t < 32; ++t) {
    const int k = t * 4 + hi * 2;
    v2f a = *(const v2f*)(xb + col * DD + k);
    wmma_row(V2, k, col, a, acc);
  }
#pragma unroll
  for (int nt = 0; nt < 8; ++nt)
#pragma unroll
    for (int v = 0; v < 8; ++v)
      out[(size_t)(b * NN + ntb + v + 8 * hi) * DD + nt * 16 + col] = acc[nt][v];
}

// ----------------------------------------------------------------------------
extern "C" void kernel_launch(void* const* d_in, const int* in_sizes, int n_in,
                              void* d_out, int out_size, void* d_ws, size_t ws_size,
                              hipStream_t stream) {
  const float* nf = (const float*)d_in[0];     // (B,N,D)
  const float* ef = (const float*)d_in[1];     // (B,E,D)
  const float* tgt = (const float*)d_in[2];    // (B,D)
  const int* ei = (const int*)d_in[3];         // (B,2,E) int32
  const float* W0 = (const float*)d_in[4];     // (384,128)
  const float* b0 = (const float*)d_in[5];
  const float* W1 = (const float*)d_in[6];
  const float* b1 = (const float*)d_in[7];
  const float* W2 = (const float*)d_in[8];
  const float* b2 = (const float*)d_in[9];
  const float* gamma = (const float*)d_in[10];
  const float* beta = (const float*)d_in[11];
  const float* V0 = (const float*)d_in[12];    // (384,128)
  const float* c0 = (const float*)d_in[13];
  const float* V1 = (const float*)d_in[14];
  const float* c1 = (const float*)d_in[15];
  const float* V2 = (const float*)d_in[16];
  const float* c2 = (const float*)d_in[17];

  float* out = (float*)d_out;    // (B,N,D) f32
  float* R = (float*)d_ws;       // (B,N,PF) accumulator, 2 MB

  const int rElems = BB * NN * PFF;
  zero_kernel<<<(rElems + 255) / 256, 256, 0, stream>>>(R, rElems);

  // 512 edge tiles, 4 waves (tiles) per 128-thread block
  edge_kernel<<<128, 128, 0, stream>>>(nf, ef, tgt, ei, W0, b0, W1, b1, W2, b2, R);

  // 256 node tiles, 4 waves per block
  node_kernel<<<64, 128, 0, stream>>>(nf, tgt, gamma, beta, V0, c0, V1, c1, V2, c2, R, out);
}